// SparseNonzeroAvgUnpooling_91053306675404
// MI455X (gfx1250) — compile-verified
//
#include <hip/hip_runtime.h>
#include <stdint.h>

// Sparse nonzero-average unpooling for MI455X (gfx1250).
// Memory-bound: ~0.2 GFLOP vs ~0.6 GB of traffic -> atomic-scatter design.
// Pass 1: u32 atomic histogram of out_map (counts), int4-vectorized.
// Pass 2: scatter in_feat[in_map[p]] * (1/count[out_map[p]]) with native
//         global_atomic_add_f32; map entries staged to LDS via per-lane
//         GLOBAL_LOAD_ASYNC_TO_LDS_B128 (ASYNCcnt path).

#define C_CH   64      // channels (fixed by reference)
#define TPB    256     // 8 wave32s per block
#define CHUNK  1024    // kernel-map pairs staged per block

#if __has_builtin(__builtin_amdgcn_global_load_async_to_lds_b128)
#define HAVE_ASYNC_LDS 1
#else
#define HAVE_ASYNC_LDS 0
#endif

typedef int v4i __attribute__((ext_vector_type(4)));
typedef __attribute__((address_space(1))) v4i gv4i;   // global (AS1)
typedef __attribute__((address_space(3))) v4i lv4i;   // LDS (AS3)

__device__ __forceinline__ void wait_async_zero() {
#if __has_builtin(__builtin_amdgcn_s_wait_asynccnt)
  __builtin_amdgcn_s_wait_asynccnt(0);
#else
  asm volatile("s_wait_asynccnt 0" ::: "memory");
#endif
}

__global__ __launch_bounds__(TPB) void unpool_scatter_kernel(
    const float* __restrict__ in_feat,
    const int* __restrict__ in_map,
    const int* __restrict__ out_map,
    const unsigned* __restrict__ cnts,
    float* __restrict__ out, int nnz) {
  __shared__ int s_in[CHUNK];
  __shared__ int s_out[CHUNK];

  const int tid = threadIdx.x;
  const int chunkBase = blockIdx.x * CHUNK;

#if HAVE_ASYNC_LDS
  {
    // Each lane asynchronously moves 16B (4 int32 map entries) of each map
    // array straight into LDS: 2x global_load_async_to_lds_b128 per lane.
    const int gbase = chunkBase + tid * 4;
    if (gbase < nnz) {
      __builtin_amdgcn_global_load_async_to_lds_b128(
          (gv4i*)(in_map + gbase), (lv4i*)(&s_in[tid * 4]), 0, 0);
      __builtin_amdgcn_global_load_async_to_lds_b128(
          (gv4i*)(out_map + gbase), (lv4i*)(&s_out[tid * 4]), 0, 0);
    }
    wait_async_zero();   // per-wave ASYNCcnt drain
  }
#else
  for (int k = tid; k < CHUNK; k += TPB) {
    const int p = chunkBase + k;
    if (p < nnz) { s_in[k] = in_map[p]; s_out[k] = out_map[p]; }
  }
#endif
  __syncthreads();

  // 16 lanes per pair, 4 channels (one float4) per lane; a wave32 covers
  // 2 pairs with fully coalesced 256B row reads (global_load_b128).
  const int g = tid >> 4;   // pair group within the 16-pair slice
  const int l = tid & 15;   // lane within pair -> channel quad
  #pragma unroll 4
  for (int sub = 0; sub < CHUNK / 16; ++sub) {
    const int pl = sub * 16 + g;
    const int p  = chunkBase + pl;
    if (p < nnz) {
      const int im = s_in[pl];
      const int om = s_out[pl];
      const unsigned c = cnts[om];
      const float inv = 1.0f / (float)(c ? c : 1u);
      const float4 v = *(const float4*)(in_feat + (size_t)im * C_CH + l * 4);
      float* op = out + (size_t)om * C_CH + l * 4;
      atomicAdd(op + 0, v.x * inv);   // -> global_atomic_add_f32
      atomicAdd(op + 1, v.y * inv);
      atomicAdd(op + 2, v.z * inv);
      atomicAdd(op + 3, v.w * inv);
    }
  }
}

__global__ __launch_bounds__(TPB) void unpool_count_kernel(
    const int* __restrict__ out_map, unsigned* __restrict__ cnts, int nnz) {
  const int base = (blockIdx.x * TPB + threadIdx.x) * 4;
  if (base + 3 < nnz) {
    // Wide path: one global_load_b128 feeding four u32 atomics.
    const v4i j = *(const v4i*)(out_map + base);
    atomicAdd(&cnts[j.x], 1u);
    atomicAdd(&cnts[j.y], 1u);
    atomicAdd(&cnts[j.z], 1u);
    atomicAdd(&cnts[j.w], 1u);
  } else {
    for (int p = base; p < nnz; ++p) atomicAdd(&cnts[out_map[p]], 1u);
  }
}

extern "C" void kernel_launch(void* const* d_in, const int* in_sizes, int n_in,
                              void* d_out, int out_size, void* d_ws, size_t ws_size,
                              hipStream_t stream) {
  const float* in_feat = (const float*)d_in[0];
  const int*   in_map  = (const int*)d_in[1];
  const int*   out_map = (const int*)d_in[2];
  const int    nnz     = in_sizes[1];
  const int    n_out   = out_size / C_CH;

  float*    out  = (float*)d_out;
  unsigned* cnts = (unsigned*)d_ws;   // n_out u32 counters (3.2 MB << ws)

  // Zero accumulators (graph-capturable memset nodes).
  (void)hipMemsetAsync(d_out, 0, (size_t)out_size * sizeof(float), stream);
  (void)hipMemsetAsync(d_ws, 0, (size_t)n_out * sizeof(unsigned), stream);

  const int cgrid = (nnz / 4 + TPB - 1) / TPB + 1;   // 4 entries per thread
  unpool_count_kernel<<<cgrid, TPB, 0, stream>>>(out_map, cnts, nnz);

  const int sgrid = (nnz + CHUNK - 1) / CHUNK;
  unpool_scatter_kernel<<<sgrid, TPB, 0, stream>>>(in_feat, in_map, out_map,
                                                   cnts, out, nnz);
}